// RelativeLoss_8529805050204
// MI455X (gfx1250) — compile-verified
//
#include <hip/hip_runtime.h>

// -------- CDNA5 (gfx1250) RelativeLoss: segment-binning + pairwise KL --------
// Pass 1: LDS-binned histogram (cnt, sum(gt), sum(sigmoid(pt))) per sample,
//         staged through LDS with gfx1250 async global->LDS loads (ASYNCcnt
//         double-buffering; proven active on this toolchain). ~100 MB streamed
//         -> ~4.3 us at 23.3 TB/s is the floor for this pass.
// Pass 2: tiled 1024x1024 pair loop; 2x v_log_f32 per pair (TRANS-bound);
//         accumulation offloaded to the matrix pipe via
//         V_WMMA_F32_16X16X4_F32 with an all-ones B (D = rowsum(A) + C), so
//         the VALU/TRANS pipes stay free for the logs. Branchless select keeps
//         EXEC all-1s (WMMA requirement) and avoids exec save/restore churn.
// Normalization sums telescope to S^2/2 exactly, so they fold into a constant
// scale and the log(Sg/Sp) cross term vanishes.

#define NUM_SP 1024
#define NSAMP  32
#define NPIX   (512 * 512)          // pixels per sample
#define BLKS_PER_SAMPLE 32          // pass-1 blocks per sample
#define PIX_PER_BLOCK   (NPIX / BLKS_PER_SAMPLE)   // 8192
#define PIX_PER_WAVE    (PIX_PER_BLOCK / 8)        // 1024 (8 waves/block)
#define WAVE_ITERS      (PIX_PER_WAVE / 32)        // 32

typedef float v2f __attribute__((ext_vector_type(2)));
typedef float v8f __attribute__((ext_vector_type(8)));

#define AS1 __attribute__((address_space(1)))
#define AS3 __attribute__((address_space(3)))

#if defined(__has_builtin)
#if __has_builtin(__builtin_amdgcn_global_load_async_to_lds_b32) && \
    __has_builtin(__builtin_amdgcn_s_wait_asynccnt)
#define HAVE_ASYNC 1
#endif
#endif
#ifndef HAVE_ASYNC
#define HAVE_ASYNC 0
#endif

__device__ __forceinline__ float fast_sigmoid(float x) {
  // 1 / (1 + exp2(-x * log2(e)))  -> v_exp_f32 + v_rcp_f32
  return __builtin_amdgcn_rcpf(1.0f + __builtin_amdgcn_exp2f(-1.44269504f * x));
}

// ---------------------------------------------------------------------------
// Pass 1: per-sample binning. Grid: NSAMP * BLKS_PER_SAMPLE blocks, 256 thr.
// bins layout (floats): [0..32K) cnt, [32K..64K) sum_gt, [64K..96K) sum_pt
// ---------------------------------------------------------------------------
__global__ __launch_bounds__(256) void bin_kernel(
    const float* __restrict__ pt, const float* __restrict__ gt,
    const int* __restrict__ sp, float* __restrict__ bins) {
  __shared__ float s_cnt[NUM_SP];
  __shared__ float s_gt[NUM_SP];
  __shared__ float s_pt[NUM_SP];
#if HAVE_ASYNC
  __shared__ int s_stage[2][8][3][32];  // [stage][wave][sp|gt|pt][lane]
#endif
  const int tid  = threadIdx.x;
  const int wave = tid >> 5;
  const int lane = tid & 31;

  for (int k = tid; k < NUM_SP; k += 256) {
    s_cnt[k] = 0.0f; s_gt[k] = 0.0f; s_pt[k] = 0.0f;
  }
  __syncthreads();

  const int sample = (int)blockIdx.x / BLKS_PER_SAMPLE;
  const int chunk  = (int)blockIdx.x % BLKS_PER_SAMPLE;
  const int base   = sample * NPIX + chunk * PIX_PER_BLOCK + wave * PIX_PER_WAVE + lane;

#if HAVE_ASYNC
  // Double-buffered async global->LDS staging (wave-local; ASYNCcnt ordered).
  {
    const int idx = base;
    __builtin_amdgcn_global_load_async_to_lds_b32(
        (AS1 int*)(sp + idx), (AS3 int*)&s_stage[0][wave][0][lane], 0, 0);
    __builtin_amdgcn_global_load_async_to_lds_b32(
        (AS1 int*)(gt + idx), (AS3 int*)&s_stage[0][wave][1][lane], 0, 0);
    __builtin_amdgcn_global_load_async_to_lds_b32(
        (AS1 int*)(pt + idx), (AS3 int*)&s_stage[0][wave][2][lane], 0, 0);
  }
  for (int i = 0; i < WAVE_ITERS; ++i) {
    const int cur = i & 1;
    if (i + 1 < WAVE_ITERS) {
      const int nxt = cur ^ 1;
      const int idx = base + (i + 1) * 32;
      __builtin_amdgcn_global_load_async_to_lds_b32(
          (AS1 int*)(sp + idx), (AS3 int*)&s_stage[nxt][wave][0][lane], 0, 0);
      __builtin_amdgcn_global_load_async_to_lds_b32(
          (AS1 int*)(gt + idx), (AS3 int*)&s_stage[nxt][wave][1][lane], 0, 0);
      __builtin_amdgcn_global_load_async_to_lds_b32(
          (AS1 int*)(pt + idx), (AS3 int*)&s_stage[nxt][wave][2][lane], 0, 0);
      __builtin_amdgcn_s_wait_asynccnt(3);  // stage `cur` fully landed
    } else {
      __builtin_amdgcn_s_wait_asynccnt(0);
    }
    const int   lbl = s_stage[cur][wave][0][lane];
    const float g   = __int_as_float(s_stage[cur][wave][1][lane]);
    const float x   = __int_as_float(s_stage[cur][wave][2][lane]);
    const float sig = fast_sigmoid(x);
    atomicAdd(&s_cnt[lbl], 1.0f);   // ds_add_f32
    atomicAdd(&s_gt[lbl],  g);
    atomicAdd(&s_pt[lbl],  sig);
  }
#else
  for (int i = 0; i < WAVE_ITERS; ++i) {
    const int idx = base + i * 32;
    const int   lbl = sp[idx];
    const float g   = gt[idx];
    const float sig = fast_sigmoid(pt[idx]);
    atomicAdd(&s_cnt[lbl], 1.0f);
    atomicAdd(&s_gt[lbl],  g);
    atomicAdd(&s_pt[lbl],  sig);
  }
#endif

  __syncthreads();
  float* cnt_g = bins + 0 * NSAMP * NUM_SP + sample * NUM_SP;
  float* gts_g = bins + 1 * NSAMP * NUM_SP + sample * NUM_SP;
  float* pts_g = bins + 2 * NSAMP * NUM_SP + sample * NUM_SP;
  for (int k = tid; k < NUM_SP; k += 256) {
    atomicAdd(&cnt_g[k], s_cnt[k]);
    atomicAdd(&gts_g[k], s_gt[k]);
    atomicAdd(&pts_g[k], s_pt[k]);
  }
}

// ---------------------------------------------------------------------------
// Pass 2: pairwise KL. Grid: (16 row-tiles of 64, 32 samples), 256 threads.
// Each wave owns 8 rows x 1024 cols: row means hoisted to registers, column
// means read as ds_load_b64 pairs, 64 fresh terms per V_WMMA_F32_16X16X4_F32.
// ---------------------------------------------------------------------------
__global__ __launch_bounds__(256) void pair_kl_kernel(
    const float* __restrict__ bins, float* __restrict__ out) {
  __shared__ float s_gm[NUM_SP];
  __shared__ float s_pm[NUM_SP];
  __shared__ float s_red[256];

  const int tid    = threadIdx.x;
  const int sample = blockIdx.y;
  const int a0     = blockIdx.x * 64;

  const float* cnt_g = bins + 0 * NSAMP * NUM_SP + sample * NUM_SP;
  const float* gts_g = bins + 1 * NSAMP * NUM_SP + sample * NUM_SP;
  const float* pts_g = bins + 2 * NSAMP * NUM_SP + sample * NUM_SP;
  for (int k = tid; k < NUM_SP; k += 256) {
    const float c  = fmaxf(cnt_g[k], 1.0f);      // sp_len[sp_len==0] = 1
    const float rc = __builtin_amdgcn_rcpf(c);
    s_gm[k] = gts_g[k] * rc;
    s_pm[k] = pts_g[k] * rc;
  }
  __syncthreads();

  const int wave = tid >> 5;
  const int lane = tid & 31;

  v8f acc = {};                       // 16x16 f32 accumulator (8 VGPRs)
  v2f ones; ones.x = 1.0f; ones.y = 1.0f;   // B = 4x16 all ones

  const float LN2  = 0.69314718055994531f;
  const float TINY = 1.0e-37f;        // clamp for branchless log (masked anyway)

  for (int rr = 0; rr < 8; ++rr) {
    const int row  = a0 + wave * 8 + rr;       // each wave owns 8 rows
    const float ga = s_gm[row];
    const float pa = s_pm[row];
#pragma unroll 4
    for (int ii = 0; ii < 16; ++ii) {
      const int col = ii * 64 + lane * 2;      // even -> 8B-aligned vector loads
      const v2f gc = *(const v2f*)&s_gm[col];  // ds_load_b64
      const v2f pc = *(const v2f*)&s_pm[col];  // ds_load_b64

      const float r0 = (ga - gc.x + 1.0f) * 0.5f;
      const float q0 = (pa - pc.x + 1.0f) * 0.5f;
      const float r1 = (ga - gc.y + 1.0f) * 0.5f;
      const float q1 = (pa - pc.y + 1.0f) * 0.5f;

      // r * (ln r - ln q); branchless: clamp log arg, mask with cndmask.
      const float l0 = __builtin_amdgcn_logf(fmaxf(r0, TINY)) - __builtin_amdgcn_logf(q0);
      const float l1 = __builtin_amdgcn_logf(fmaxf(r1, TINY)) - __builtin_amdgcn_logf(q1);
      float t0 = r0 * LN2 * l0;
      float t1 = r1 * LN2 * l1;
      t0 = (r0 > 0.0f) ? t0 : 0.0f;
      t1 = (r1 > 0.0f) ? t1 : 0.0f;

      v2f a; a.x = t0; a.y = t1;      // A = 16x4 tile: 64 fresh terms / wave
      acc = __builtin_amdgcn_wmma_f32_16x16x4_f32(
          /*neg_a=*/false, a, /*neg_b=*/false, ones,
          /*c_mod=*/(short)0, acc, /*reuse_a=*/false, /*reuse_b=*/false);
    }
  }

  // Each column of D holds the same row-sums (B was all ones) -> full-matrix
  // sum = 16 x true sum of accumulated terms.
  float ls = acc[0] + acc[1] + acc[2] + acc[3] + acc[4] + acc[5] + acc[6] + acc[7];
  s_red[tid] = ls;
  __syncthreads();
  for (int off = 128; off > 0; off >>= 1) {
    if (tid < off) s_red[tid] += s_red[tid + off];
    __syncthreads();
  }
  if (tid == 0) {
    // Sg = Sp = S^2/2 exactly (telescoping), so loss_b = (2/S^2) * sum(terms);
    // final = sum_b loss_b / B. Divide by 16 for the ones-B replication.
    const float SCALE =
        (1.0f / 16.0f) * (2.0f / ((float)NUM_SP * (float)NUM_SP)) / (float)NSAMP;
    atomicAdd(out, s_red[0] * SCALE);
  }
}

// ---------------------------------------------------------------------------
extern "C" void kernel_launch(void* const* d_in, const int* in_sizes, int n_in,
                              void* d_out, int out_size, void* d_ws, size_t ws_size,
                              hipStream_t stream) {
  (void)in_sizes; (void)n_in; (void)out_size; (void)ws_size;
  const float* pt = (const float*)d_in[0];
  const float* gt = (const float*)d_in[1];
  const int*   sp = (const int*)d_in[2];
  float* bins = (float*)d_ws;           // 3 * 32 * 1024 floats = 384 KB

  (void)hipMemsetAsync(bins, 0, (size_t)3 * NSAMP * NUM_SP * sizeof(float), stream);
  (void)hipMemsetAsync(d_out, 0, sizeof(float), stream);

  bin_kernel<<<dim3(NSAMP * BLKS_PER_SAMPLE), dim3(256), 0, stream>>>(pt, gt, sp, bins);
  pair_kl_kernel<<<dim3(NUM_SP / 64, NSAMP), dim3(256), 0, stream>>>(bins, (float*)d_out);
}